// GaussianMultiViewMerge_13924283973708
// MI455X (gfx1250) — compile-verified
//
#include <hip/hip_runtime.h>
#include <hip/hip_bf16.h>
#include <string.h>

// ---------------------------------------------------------------------------
// GaussianMultiViewMerge for MI455X (gfx1250, wave32).
// Memory-bound segmented softmax merge. Segment tables (~14.7MB) stay in L2
// (192MB); element data streamed ~2.3x => ~510MB HBM traffic, ~22us @23.3TB/s.
// WMMA used where it is numerically exact: quaternion norms via the diagonal
// of R*R^T with V_WMMA_F32_16X16X4_F32 (f32 in/out, RNE).
// ---------------------------------------------------------------------------

#define MAXID     8192
#define BT        4
#define SEG_DUMMY (BT * MAXID)     // 32768
#define NSEG      (SEG_DUMMY + 1)  // 32769
#define NCH       106              // channels that take the weighted mean
#define THRV      2.0f
#define DUPF      0.05f
#define NEGV      (-1e30f)

typedef __attribute__((ext_vector_type(2))) float v2f;
typedef __attribute__((ext_vector_type(8))) float v8f;

// Order-preserving float<->uint mapping so segment max works with u32 atomics
// (deterministic, exact bit recovery of the max).
__device__ __forceinline__ unsigned ford(float f) {
  unsigned u = __float_as_uint(f);
  return u ^ ((u >> 31) ? 0xFFFFFFFFu : 0x80000000u);
}
__device__ __forceinline__ float uford(unsigned x) {
  unsigned u = (x & 0x80000000u) ? (x ^ 0x80000000u) : ~x;
  return __uint_as_float(u);
}

__global__ void k_fill_u32(unsigned* __restrict__ p, unsigned v, long long n) {
  for (long long t = (long long)blockIdx.x * blockDim.x + threadIdx.x; t < n;
       t += (long long)gridDim.x * blockDim.x)
    p[t] = v;
}

// Pass 1: segment id, valid count, segment max of keep0 over valid elements.
__global__ void k_seg_cnt_mx1(const int* __restrict__ gid,
                              const float* __restrict__ keep,
                              int* __restrict__ seg, unsigned* __restrict__ cnt,
                              unsigned* __restrict__ mx1, int M, int vhw) {
  int i = blockIdx.x * blockDim.x + threadIdx.x;
  if (i >= M) return;
  int id = gid[i];
  int s = SEG_DUMMY;
  if (id >= 0) {
    s = (i / vhw) * MAXID + id;
    atomicAdd(&cnt[s], 1u);
    atomicMax(&mx1[s], ford(keep[i]));
  }
  seg[i] = s;
}

// Pass 2: softmax-1 numerators: d1 += e1, mc1 += center*e1.
__global__ void k_e1_mc1(const float* __restrict__ keep,
                         const float* __restrict__ cen,
                         const int* __restrict__ seg,
                         const unsigned* __restrict__ mx1,
                         float* __restrict__ d1, float* __restrict__ mc1,
                         int M) {
  int i = blockIdx.x * blockDim.x + threadIdx.x;
  if (i >= M) return;
  int s = seg[i];
  if (s == SEG_DUMMY) return;
  float e = expf(keep[i] - uford(mx1[s]));
  atomicAdd(&d1[s], e);
  atomicAdd(&mc1[(long long)s * 3 + 0], cen[(long long)i * 3 + 0] * e);
  atomicAdd(&mc1[(long long)s * 3 + 1], cen[(long long)i * 3 + 1] * e);
  atomicAdd(&mc1[(long long)s * 3 + 2], cen[(long long)i * 3 + 2] * e);
}

__global__ void k_mc1_final(float* __restrict__ mc1,
                            const float* __restrict__ d1, int n3) {
  int t = blockIdx.x * blockDim.x + threadIdx.x;
  if (t >= n3) return;
  float d = d1[t / 3];
  if (d > 0.0f) mc1[t] /= d;
}

// Pass 3: active mask (elig & dist<=THR), segment max of keep0 over active.
__global__ void k_active_mx2(const float* __restrict__ keep,
                             const float* __restrict__ cen,
                             const int* __restrict__ seg,
                             const unsigned* __restrict__ cnt,
                             const float* __restrict__ mc1,
                             unsigned char* __restrict__ act,
                             unsigned* __restrict__ mx2, int M) {
  int i = blockIdx.x * blockDim.x + threadIdx.x;
  if (i >= M) return;
  int s = seg[i];
  unsigned char a = 0;
  if (s != SEG_DUMMY && cnt[s] >= 2u) {
    float dx = cen[(long long)i * 3 + 0] - mc1[(long long)s * 3 + 0];
    float dy = cen[(long long)i * 3 + 1] - mc1[(long long)s * 3 + 1];
    float dz = cen[(long long)i * 3 + 2] - mc1[(long long)s * 3 + 2];
    float dist = sqrtf(dx * dx + dy * dy + dz * dz);  // match jnp.linalg.norm
    if (dist <= THRV) {
      a = 1;
      atomicMax(&mx2[s], ford(keep[i]));
    }
  }
  act[i] = a;
}

// Pass 4: e2, d2, representative (min index among keep0 == segment max).
__global__ void k_e2_d2_minidx(const float* __restrict__ keep,
                               const int* __restrict__ seg,
                               const unsigned char* __restrict__ act,
                               const unsigned* __restrict__ mx2,
                               float* __restrict__ e2, float* __restrict__ d2,
                               unsigned* __restrict__ minidx, int M) {
  int i = blockIdx.x * blockDim.x + threadIdx.x;
  if (i >= M) return;
  if (!act[i]) {
    e2[i] = 0.0f;
    return;
  }
  int s = seg[i];
  float k0 = keep[i];
  float m = uford(mx2[s]);
  float e = expf(k0 - m);
  e2[i] = e;
  atomicAdd(&d2[s], e);
  if (k0 >= m) atomicMin(&minidx[s], (unsigned)i);
}

// Channel-parallel weighted-sum scatter (coalesced reads, L2-resident atomics).
__global__ void k_accum(const float* __restrict__ in, float* __restrict__ msum,
                        const int* __restrict__ seg,
                        const unsigned char* __restrict__ act,
                        const float* __restrict__ e2, int C, int base, int M) {
  long long n = (long long)M * C;
  for (long long t = (long long)blockIdx.x * blockDim.x + threadIdx.x; t < n;
       t += (long long)gridDim.x * blockDim.x) {
    long long i = t / C;
    int c = (int)(t - i * C);
    if (!act[i]) continue;
    atomicAdd(&msum[(long long)seg[i] * NCH + base + c], in[t] * e2[i]);
  }
}

__global__ void k_mean_final(float* __restrict__ msum,
                             const float* __restrict__ d2, long long n) {
  for (long long t = (long long)blockIdx.x * blockDim.x + threadIdx.x; t < n;
       t += (long long)gridDim.x * blockDim.x) {
    float d = d2[t / NCH];
    if (d > 0.0f) msum[t] /= d;
  }
}

// Quaternion renormalization via WMMA: one wave handles 16 segments.
// A (16x4 f32) = R, and per the ISA register layouts B = R^T lands in the SAME
// registers as A, so D = A*A^T gives ||rs||^2 on its diagonal. Full f32, exact.
__global__ void k_rot_norm_wmma(float* __restrict__ msum, int S) {
  int lane = threadIdx.x & 31;
  int wave = blockIdx.x * (blockDim.x >> 5) + (threadIdx.x >> 5);
  int s0 = wave * 16;
  if (s0 >= S) return;  // wave-uniform exit: EXEC stays all-1s below

  int mrow = lane & 15;
  int s = s0 + mrow;
  int sc = (s < S) ? s : (S - 1);  // clamp loads; writes are predicated
  const float* rp = msum + (long long)sc * NCH + 74;  // rotation base

  bool hi = lane >= 16;
  v2f a;
  a.x = rp[hi ? 2 : 0];  // A layout: lanes 0-15 hold K0/K1, lanes 16-31 K2/K3
  a.y = rp[hi ? 3 : 1];

  v8f c = {};
  // D = A * A^T (+0): v_wmma_f32_16x16x4_f32
  c = __builtin_amdgcn_wmma_f32_16x16x4_f32(false, a, false, a, (short)0, c,
                                            false, false);

  // Diagonal D[m][m]: m<8 -> vgpr m @ lane m ; m>=8 -> vgpr m-8 @ lane m+16.
  float diag = 0.0f;
  int ci = (lane < 8) ? lane : ((lane >= 24) ? (lane - 24) : 0);
  if (lane < 8 || lane >= 24) diag = c[ci];
  int src = (mrow < 8) ? mrow : (mrow + 16);
  float n2 = __shfl(diag, src, 32);

  if (lane < 16 && s < S) {
    float rinv = 1.0f / fmaxf(sqrtf(n2), 1e-12f);
    float* wp = msum + (long long)s * NCH + 74;
    float r0 = rp[0], r1 = rp[1], r2 = rp[2], r3 = rp[3];
    wp[0] = r0 * rinv;
    wp[1] = r1 * rinv;
    wp[2] = r2 * rinv;
    wp[3] = r3 * rinv;
  }
}

// Output gather: mean where active, passthrough otherwise.
__global__ void k_out_copy(const float* __restrict__ in, float* __restrict__ out,
                           const float* __restrict__ msum,
                           const int* __restrict__ seg,
                           const unsigned char* __restrict__ act, int C,
                           int base, int M) {
  long long n = (long long)M * C;
  for (long long t = (long long)blockIdx.x * blockDim.x + threadIdx.x; t < n;
       t += (long long)gridDim.x * blockDim.x) {
    if (t + 8192 < n) __builtin_prefetch(&in[t + 8192], 0, 0);
    long long i = t / C;
    int c = (int)(t - i * C);
    out[t] = act[i] ? msum[(long long)seg[i] * NCH + base + c] : in[t];
  }
}

// Opacity / keep_score: mean * (is_rep ? 1 : DUP). keep mean == mx2 exactly.
__global__ void k_out_factor(const float* __restrict__ in,
                             float* __restrict__ out,
                             const float* __restrict__ msum,
                             const unsigned* __restrict__ mx2,
                             const int* __restrict__ seg,
                             const unsigned char* __restrict__ act,
                             const unsigned* __restrict__ minidx, int base,
                             int M, int use_mx2) {
  int i = blockIdx.x * blockDim.x + threadIdx.x;
  if (i >= M) return;
  if (!act[i]) {
    out[i] = in[i];
    return;
  }
  int s = seg[i];
  float mean = use_mx2 ? uford(mx2[s]) : msum[(long long)s * NCH + base];
  float fac = ((unsigned)i == minidx[s]) ? 1.0f : DUPF;
  out[i] = mean * fac;
}

static inline unsigned ord_host(float f) {
  unsigned u;
  memcpy(&u, &f, 4);
  return u ^ ((u >> 31) ? 0xFFFFFFFFu : 0x80000000u);
}

extern "C" void kernel_launch(void* const* d_in, const int* in_sizes, int n_in,
                              void* d_out, int out_size, void* d_ws,
                              size_t ws_size, hipStream_t stream) {
  const float* dense  = (const float*)d_in[0];
  const float* cen    = (const float*)d_in[1];
  const float* off    = (const float*)d_in[2];
  const float* opa    = (const float*)d_in[3];
  const float* scl    = (const float*)d_in[4];
  const float* rot    = (const float*)d_in[5];
  const float* fdc    = (const float*)d_in[6];
  const float* keep   = (const float*)d_in[7];
  const float* inst   = (const float*)d_in[8];
  const float* mot    = (const float*)d_in[9];
  const float* dyn    = (const float*)d_in[10];
  const int*   gid    = (const int*)d_in[11];
  float* out = (float*)d_out;

  const int M = in_sizes[11];     // number of elements
  const int vhw = M / BT;         // V*H*W

  // Workspace carve-up (all 4B aligned; ~18.6 MB total).
  char* w = (char*)d_ws;
  int*      seg    = (int*)w;           w += (size_t)M * 4;
  float*    e2     = (float*)w;         w += (size_t)M * 4;
  unsigned* mx1    = (unsigned*)w;      w += (size_t)NSEG * 4;
  unsigned* mx2    = (unsigned*)w;      w += (size_t)NSEG * 4;
  unsigned* minidx = (unsigned*)w;      w += (size_t)NSEG * 4;
  unsigned* cnt    = (unsigned*)w;      w += (size_t)NSEG * 4;  // zero region starts
  float*    d1     = (float*)w;         w += (size_t)NSEG * 4;
  float*    d2     = (float*)w;         w += (size_t)NSEG * 4;
  float*    mc1    = (float*)w;         w += (size_t)NSEG * 12;
  float*    msum   = (float*)w;         w += (size_t)NSEG * NCH * 4;
  unsigned char* act = (unsigned char*)w;

  const int TB = 256;
  auto cdiv = [](long long a, long long b) { return (unsigned)((a + b - 1) / b); };

  // Init: zeros for cnt,d1,d2,mc1,msum (contiguous), ord(NEG) for mx1,mx2,
  // M for minidx.
  long long nz = (long long)NSEG * (1 + 1 + 1 + 3 + NCH);
  k_fill_u32<<<cdiv(nz, TB), TB, 0, stream>>>(cnt, 0u, nz);
  k_fill_u32<<<cdiv(2LL * NSEG, TB), TB, 0, stream>>>(mx1, ord_host(NEGV),
                                                      2LL * NSEG);
  k_fill_u32<<<cdiv((long long)NSEG, TB), TB, 0, stream>>>(minidx, (unsigned)M,
                                                           (long long)NSEG);

  k_seg_cnt_mx1<<<cdiv(M, TB), TB, 0, stream>>>(gid, keep, seg, cnt, mx1, M, vhw);
  k_e1_mc1<<<cdiv(M, TB), TB, 0, stream>>>(keep, cen, seg, mx1, d1, mc1, M);
  k_mc1_final<<<cdiv(3LL * NSEG, TB), TB, 0, stream>>>(mc1, d1, 3 * NSEG);
  k_active_mx2<<<cdiv(M, TB), TB, 0, stream>>>(keep, cen, seg, cnt, mc1, act,
                                               mx2, M);
  k_e2_d2_minidx<<<cdiv(M, TB), TB, 0, stream>>>(keep, seg, act, mx2, e2, d2,
                                                 minidx, M);

  // Weighted sums for all 106 mean channels (base offsets into msum rows).
  struct T { const float* p; int C; int base; };
  const T tens[10] = {{dense, 64, 0},  {cen, 3, 64},  {off, 3, 67},
                      {opa, 1, 70},    {scl, 3, 71},  {rot, 4, 74},
                      {fdc, 3, 78},    {inst, 16, 81},{mot, 8, 97},
                      {dyn, 1, 105}};
  for (int k = 0; k < 10; ++k)
    k_accum<<<cdiv((long long)M * tens[k].C, TB), TB, 0, stream>>>(
        tens[k].p, msum, seg, act, e2, tens[k].C, tens[k].base, M);

  k_mean_final<<<cdiv((long long)NSEG * NCH, TB), TB, 0, stream>>>(
      msum, d2, (long long)NSEG * NCH);

  // WMMA quaternion renormalization: 16 segments per wave, 8 waves per block.
  k_rot_norm_wmma<<<cdiv((long long)NSEG, 16LL * 8LL), 256, 0, stream>>>(msum,
                                                                         NSEG);

  // Outputs (return-order concatenation; offsets in floats).
  const long long Mo = M;
  struct O { const float* in; long long ooff; int C; int base; };
  const O outs[9] = {{dense, 0, 64, 0},       {cen, 64 * Mo, 3, 64},
                     {off, 67 * Mo, 3, 67},   {scl, 71 * Mo, 3, 71},
                     {rot, 74 * Mo, 4, 74},   {fdc, 78 * Mo, 3, 78},
                     {inst, 82 * Mo, 16, 81}, {mot, 98 * Mo, 8, 97},
                     {dyn, 106 * Mo, 1, 105}};
  for (int k = 0; k < 9; ++k)
    k_out_copy<<<cdiv((long long)M * outs[k].C, TB), TB, 0, stream>>>(
        outs[k].in, out + outs[k].ooff, msum, seg, act, outs[k].C, outs[k].base,
        M);

  // opacity (mean from msum) and keep_score (mean == mx2), both with factor.
  k_out_factor<<<cdiv(M, TB), TB, 0, stream>>>(opa, out + 70 * Mo, msum, mx2,
                                               seg, act, minidx, 70, M, 0);
  k_out_factor<<<cdiv(M, TB), TB, 0, stream>>>(keep, out + 81 * Mo, msum, mx2,
                                               seg, act, minidx, 0, M, 1);
}